// IntSoftmax_70909910057500
// MI455X (gfx1250) — compile-verified
//
#include <hip/hip_runtime.h>
#include <stdint.h>

// ---------------------------------------------------------------------------
// I-BERT integer softmax, [1,8,2048,2048] f32, dim=3.
// Kernel 1: global amax(|x|) reduction  (134 MB read, HBM bound)
// Kernel 2: one workgroup per 2048-elem row (134 MB read — L2 resident in
//           MI455X's 192MB L2 — + 134 MB write)
// CDNA5 paths: global_load_async_to_lds_b128 (ASYNCcnt) row staging,
//              v_wmma_f32_16x16x4_f32 wave-sum reduction, wave32 shuffles.
// All row-invariant divisions hoisted to one v_rcp_f32 each; the per-element
// loop is pure mul/fma/floor/ldexp so VALU stays out of the way of the
// bandwidth-bound memory pipeline.
// ---------------------------------------------------------------------------

#define ROW_LEN 2048
#define BLOCK   256   // 8 waves of 32; 8 elements per thread

typedef __attribute__((ext_vector_type(2))) float v2f;
typedef __attribute__((ext_vector_type(8))) float v8f;

__device__ __forceinline__ float wave_reduce_max(float m) {
#pragma unroll
  for (int i = 16; i > 0; i >>= 1)
    m = fmaxf(m, __shfl_xor(m, i, 32));
  return m;
}

// Wave32 sum via one V_WMMA_F32_16X16X4_F32.
// A(16x4): lane L<16 -> {A[L][0],A[L][1]} = {p,0}; lane L>=16 -> {A[L-16][2],A[L-16][3]} = {p,0}
// B(4x16) = ones  =>  D[m][n] = p_m + p_{m+16}
// D(16x16): VGPR j = D[j][lane] (lanes 0-15) / D[j+8][lane-16] (lanes 16-31)
// sum of the 8 D regs gives half-sums; shfl_xor(16) completes the reduction.
__device__ __forceinline__ float wave_reduce_sum_wmma(float p) {
  v2f a; a.x = p;    a.y = 0.0f;
  v2f b; b.x = 1.0f; b.y = 1.0f;
  v8f c = {};
  v8f d = __builtin_amdgcn_wmma_f32_16x16x4_f32(
      /*neg_a=*/false, a, /*neg_b=*/false, b,
      /*c_mod=*/(short)0, c, /*reuse_a=*/false, /*reuse_b=*/false);
  float s = d[0] + d[1] + d[2] + d[3] + d[4] + d[5] + d[6] + d[7];
  s += __shfl_xor(s, 16, 32);
  return s;
}

// -------------------------- kernel 0: init ws ------------------------------
__global__ void init_amax(unsigned* amax_bits) { amax_bits[0] = 0u; }

// -------------------------- kernel 1: amax(|x|) ----------------------------
__global__ void __launch_bounds__(256)
amax_kernel(const float4* __restrict__ x4, unsigned* __restrict__ amax_bits, int n4) {
  float m = 0.0f;
  const int stride = gridDim.x * blockDim.x;
  for (int i = blockIdx.x * blockDim.x + threadIdx.x; i < n4; i += stride) {
    const float4 v = x4[i];
    m = fmaxf(m, fmaxf(fmaxf(fabsf(v.x), fabsf(v.y)),
                       fmaxf(fabsf(v.z), fabsf(v.w))));
  }
  m = wave_reduce_max(m);
  // |x| >= 0 so raw float bits are monotonic under unsigned compare
  if ((threadIdx.x & 31) == 0) atomicMax(amax_bits, __float_as_uint(m));
}

// ---------------------- kernel 2: per-row int softmax ----------------------
__global__ void __launch_bounds__(BLOCK)
isoftmax_rows(const float* __restrict__ x, float* __restrict__ out,
              const unsigned* __restrict__ amax_bits) {
  __shared__ float lds_row[ROW_LEN];   // 8 KB row stage
  __shared__ float red_max[8];
  __shared__ float red_sum[8];

  const int t = threadIdx.x;
  const long long base = (long long)blockIdx.x * ROW_LEN;
  const float* src = x + base;

  // ---- CDNA5 async DMA: stage the 8KB row into LDS (ASYNCcnt path) ----
  {
    // low 32 bits of a generic LDS pointer == wave-relative LDS byte offset
    unsigned loff = (unsigned)(uintptr_t)&lds_row[0] + (unsigned)t * 16u;
    unsigned long long g = (unsigned long long)(uintptr_t)src
                         + (unsigned long long)t * 16ull;
    asm volatile("global_load_async_to_lds_b128 %0, %1, off"
                 :: "v"(loff), "v"(g) : "memory");
    asm volatile("global_load_async_to_lds_b128 %0, %1, off"
                 :: "v"(loff + 4096u), "v"(g + 4096ull) : "memory");
    asm volatile("s_wait_asynccnt 0" ::: "memory");
  }
  __syncthreads();

  // ---- row-invariant constants: ONE fast reciprocal each, no per-elem div
  const float amax      = __uint_as_float(amax_bits[0]);
  const float scale     = amax * (1.0f / 127.0f);   // max(|min|,|max|)/n
  const float inv_scale = __builtin_amdgcn_rcpf(scale);

  const float x0i     = floorf(-0.6931f * inv_scale);                 // < 0
  const float inv_x0i = __builtin_amdgcn_rcpf(x0i);
  const float bi      = floorf((0.96963238f / 0.35815147f) * inv_scale);
  const float ci      = floorf((1.0f / 0.35815147f) * inv_scale * inv_scale);
  const float lo      = 30.0f * x0i;

  // ---- quantize my 8 elements (kept in registers thereafter) ----
  float xi[8];
  {
    const float4 v0 = *(const float4*)&lds_row[t * 8];
    const float4 v1 = *(const float4*)&lds_row[t * 8 + 4];
    const float tmp[8] = {v0.x, v0.y, v0.z, v0.w, v1.x, v1.y, v1.z, v1.w};
#pragma unroll
    for (int i = 0; i < 8; ++i)
      xi[i] = fminf(fmaxf(rintf(tmp[i] * inv_scale), -128.0f), 127.0f);
  }

  // ---- row max ----
  float m = xi[0];
#pragma unroll
  for (int i = 1; i < 8; ++i) m = fmaxf(m, xi[i]);
  m = wave_reduce_max(m);
  if ((t & 31) == 0) red_max[t >> 5] = m;
  __syncthreads();
  float rowmax = red_max[0];
#pragma unroll
  for (int i = 1; i < 8; ++i) rowmax = fmaxf(rowmax, red_max[i]);

  // ---- integer exp (I-BERT range reduction + 2nd-order poly) ----
  float e[8];
  float psum = 0.0f;
#pragma unroll
  for (int i = 0; i < 8; ++i) {
    float v = fmaxf(xi[i] - rowmax, lo);
    float q = floorf(v * inv_x0i);           // q in [0,30]
    float r = fmaf(-x0i, q, v);              // v - x0i*q
    float z = fmaf(r + bi, r, ci);           // (r+b)*r + c
    float ev = fmaxf(floorf(ldexpf(z, 30 - (int)q)), 0.0f);  // z * 2^(30-q)
    e[i] = ev;
    psum += ev;
  }

  // ---- row sum (WMMA wave reduction + LDS cross-wave) ----
  const float ws = wave_reduce_sum_wmma(psum);
  if ((t & 31) == 0) red_sum[t >> 5] = ws;
  __syncthreads();
  float S = red_sum[0];
#pragma unroll
  for (int i = 1; i < 8; ++i) S += red_sum[i];

  // ---- normalize: factor = floor(2^32 / sum); requantize to 8 bit ----
  // one divide per thread is fine; keep it exact like the reference
  const float factor = floorf(4294967296.0f / S);
  float4 o0, o1;
  o0.x = floorf(e[0] * factor * 0x1.0p-24f) * (1.0f / 256.0f);
  o0.y = floorf(e[1] * factor * 0x1.0p-24f) * (1.0f / 256.0f);
  o0.z = floorf(e[2] * factor * 0x1.0p-24f) * (1.0f / 256.0f);
  o0.w = floorf(e[3] * factor * 0x1.0p-24f) * (1.0f / 256.0f);
  o1.x = floorf(e[4] * factor * 0x1.0p-24f) * (1.0f / 256.0f);
  o1.y = floorf(e[5] * factor * 0x1.0p-24f) * (1.0f / 256.0f);
  o1.z = floorf(e[6] * factor * 0x1.0p-24f) * (1.0f / 256.0f);
  o1.w = floorf(e[7] * factor * 0x1.0p-24f) * (1.0f / 256.0f);

  float4* op = (float4*)(out + base) + (size_t)t * 2;
  op[0] = o0;
  op[1] = o1;
}

// ------------------------------- launcher ----------------------------------
extern "C" void kernel_launch(void* const* d_in, const int* in_sizes, int n_in,
                              void* d_out, int out_size, void* d_ws, size_t ws_size,
                              hipStream_t stream) {
  const float* x   = (const float*)d_in[0];   // [1,8,2048,2048] f32
  float*       out = (float*)d_out;           // same shape, f32
  unsigned* amax_bits = (unsigned*)d_ws;      // 4 bytes of scratch

  const int n    = in_sizes[0];
  const int rows = n / ROW_LEN;

  init_amax<<<1, 1, 0, stream>>>(amax_bits);
  amax_kernel<<<4096, 256, 0, stream>>>((const float4*)x, amax_bits, n / 4);
  isoftmax_rows<<<rows, BLOCK, 0, stream>>>(x, out, amax_bits);
}